// MusicAutoregressiveWrapper_21139829031085
// MI455X (gfx1250) — compile-verified
//
#include <hip/hip_runtime.h>
#include <cstdint>
#include <cstddef>

// ---------------- problem constants (from reference) ----------------
#define Dz    8
#define Bz    4096
#define Vz    1024
#define KSEL  102        // int((1 - 0.9) * 1024)
#define NOTE  1

// ---------------- launch geometry ----------------
#define WPB   4                 // waves per block (wave32)
#define BLK   (WPB * 32)        // 128 threads
#define NBLK  2048
#define TOTW  (NBLK * WPB)      // 8192 waves
#define ROWS  (Dz * Bz)         // 32768 rows
#define RPW   (ROWS / TOTW)     // 4 rows per wave

#define NEG_INF (-__builtin_inff())

// ---------------- helpers ----------------
__device__ __forceinline__ unsigned f2key(float f) {
  unsigned v = __float_as_uint(f);
  return (v & 0x80000000u) ? ~v : (v | 0x80000000u);   // monotonic float->uint
}
__device__ __forceinline__ float key2f(unsigned k) {
  unsigned v = (k & 0x80000000u) ? (k & 0x7fffffffu) : ~k;
  return __uint_as_float(v);
}
// Force a wave-uniform 64-bit value into SGPRs for the async-load saddr.
__device__ __forceinline__ uint64_t uniform64(uint64_t x) {
  uint32_t lo = (uint32_t)__builtin_amdgcn_readfirstlane((int)(uint32_t)x);
  uint32_t hi = (uint32_t)__builtin_amdgcn_readfirstlane((int)(uint32_t)(x >> 32));
  return ((uint64_t)hi << 32) | lo;
}

// Async-copy one row (1024 logits + 1024 u floats) into this wave's LDS buffer.
// vdst = LDS byte offset (buffer base + lane*16), vaddr = lane*16.
// The 24-bit inst offset is added to BOTH the LDS and global address, so the
// u base pointer is pre-biased by -4096 bytes (u lands at LDS buffer +4096).
__device__ __forceinline__ void async_row(uint32_t vdst, uint32_t vaddr,
                                          uint64_t s_logits, uint64_t s_u_m4k) {
  // make sure prior ds reads of this buffer have fully retired
  asm volatile("s_wait_dscnt 0x0" ::: "memory");
#define AL(OFF) asm volatile("global_load_async_to_lds_b128 %0, %1, %2 offset:" OFF \
                             :: "v"(vdst), "v"(vaddr), "s"(s_logits) : "memory")
  AL("0");    AL("512");  AL("1024"); AL("1536");
  AL("2048"); AL("2560"); AL("3072"); AL("3584");
#undef AL
#define AU(OFF) asm volatile("global_load_async_to_lds_b128 %0, %1, %2 offset:" OFF \
                             :: "v"(vdst), "v"(vaddr), "s"(s_u_m4k) : "memory")
  AU("4096"); AU("4608"); AU("5120"); AU("5632");
  AU("6144"); AU("6656"); AU("7168"); AU("7680");
#undef AU
}

// ---------------- kernel 1: per-row top-k softmax + Gumbel sample ----------------
__global__ __launch_bounds__(BLK) void decode_step(
    const float* __restrict__ logits, const float* __restrict__ u,
    const int* __restrict__ cur, float* __restrict__ probs_out,
    int* __restrict__ samples_out)
{
  __shared__ __align__(16) float smem[WPB * 4096];   // per wave: 2 bufs x (1024+1024) floats
  const int lane = threadIdx.x & 31;
  const int wq   = threadIdx.x >> 5;
  const int wave = blockIdx.x * WPB + wq;

  const uint32_t ldsWave = (uint32_t)(uintptr_t)smem + (uint32_t)wq * 16384u;
  const uint32_t vaddr   = (uint32_t)lane * 16u;

  int r = wave;
  {   // prime the pipeline: row r -> buffer 0
    uint64_t sl = uniform64((uint64_t)(uintptr_t)logits + (uint64_t)r * (Vz * 4));
    uint64_t su = uniform64((uint64_t)(uintptr_t)u + (uint64_t)r * (Vz * 4) - 4096u);
    async_row(ldsWave + vaddr, vaddr, sl, su);
  }

  int p = 0;
  #pragma unroll 1
  for (int it = 0; it < RPW; ++it) {
    if (it + 1 < RPW) {
      int rn = r + TOTW;
      uint64_t sl = uniform64((uint64_t)(uintptr_t)logits + (uint64_t)rn * (Vz * 4));
      uint64_t su = uniform64((uint64_t)(uintptr_t)u + (uint64_t)rn * (Vz * 4) - 4096u);
      async_row(ldsWave + (uint32_t)(p ^ 1) * 8192u + vaddr, vaddr, sl, su);
      asm volatile("s_wait_asynccnt 0x10" ::: "memory");   // current buffer's 16 done
    } else {
      asm volatile("s_wait_asynccnt 0x0" ::: "memory");
    }

    const float4* slog4 = (const float4*)(smem + wq * 4096 + p * 2048);
    const float4* su4   = (const float4*)(smem + wq * 4096 + p * 2048 + 1024);

    const int cv  = cur[r];
    const int thr = cv > 1 ? cv : 1;       // mask v < cur  and  v == 0

    float    lgv[8][4];
    unsigned key[8][4];
    #pragma unroll
    for (int j = 0; j < 8; ++j) {
      float4 x = slog4[j * 32 + lane];
      float xs[4] = {x.x, x.y, x.z, x.w};
      int vb = j * 128 + lane * 4;
      #pragma unroll
      for (int c = 0; c < 4; ++c) {
        float lg = (vb + c < thr) ? NEG_INF : xs[c];
        lgv[j][c] = lg;
        key[j][c] = f2key(lg);
      }
    }

    // exact KSEL-th largest via bitwise radix select; wave32 ballot counting
    unsigned prefix = 0u;
    int kk = KSEL;
    for (int bit = 31; bit >= 0; --bit) {
      unsigned want = (prefix >> bit) | 1u;
      int cnt = 0;
      #pragma unroll
      for (int j = 0; j < 8; ++j)
        #pragma unroll
        for (int c = 0; c < 4; ++c)
          cnt += __popc(__builtin_amdgcn_ballot_w32((key[j][c] >> bit) == want));
      if (cnt >= kk) prefix |= (1u << bit); else kk -= cnt;
    }

    // wave max (in key space) -> stable softmax max
    unsigned mk = 0u;
    #pragma unroll
    for (int j = 0; j < 8; ++j)
      #pragma unroll
      for (int c = 0; c < 4; ++c) mk = key[j][c] > mk ? key[j][c] : mk;
    #pragma unroll
    for (int o = 16; o >= 1; o >>= 1) {
      unsigned om = (unsigned)__shfl_xor((int)mk, o, 32);
      mk = om > mk ? om : mk;
    }
    const float m = key2f(mk);

    float e[8][4];
    float s = 0.f;
    #pragma unroll
    for (int j = 0; j < 8; ++j)
      #pragma unroll
      for (int c = 0; c < 4; ++c) {
        float filt = (key[j][c] >= prefix) ? lgv[j][c] : NEG_INF;
        float ev = __expf(filt - m);      // exp(-inf) -> 0
        e[j][c] = ev;
        s += ev;
      }
    #pragma unroll
    for (int o = 16; o >= 1; o >>= 1) s += __shfl_xor(s, o, 32);
    const float logZ = m + __logf(s);
    const float rcp  = 1.0f / s;

    float* orow = probs_out + (size_t)r * Vz;
    #pragma unroll
    for (int j = 0; j < 8; ++j) {
      float4 pr;
      pr.x = e[j][0] * rcp; pr.y = e[j][1] * rcp;
      pr.z = e[j][2] * rcp; pr.w = e[j][3] * rcp;
      ((float4*)orow)[j * 32 + lane] = pr;
    }

    // Gumbel-max sampling: argmax(logp + gumbel), ties -> lowest index
    float bestv = NEG_INF; int besti = 0;
    #pragma unroll
    for (int j = 0; j < 8; ++j) {
      float4 uu = su4[j * 32 + lane];
      float us[4] = {uu.x, uu.y, uu.z, uu.w};
      int vb = j * 128 + lane * 4;
      #pragma unroll
      for (int c = 0; c < 4; ++c) {
        float g  = -__logf(-__logf(us[c] + 1e-20f) + 1e-20f);
        float filt = (key[j][c] >= prefix) ? lgv[j][c] : NEG_INF;
        float sc = (filt - logZ) + g;
        if (sc > bestv) { bestv = sc; besti = vb + c; }
      }
    }
    #pragma unroll
    for (int o = 16; o >= 1; o >>= 1) {
      float ov = __shfl_xor(bestv, o, 32);
      int   oi = __shfl_xor(besti, o, 32);
      if (ov > bestv || (ov == bestv && oi < besti)) { bestv = ov; besti = oi; }
    }
    if (lane == 0) samples_out[r] = besti;

    p ^= 1;
    r += TOTW;
  }
}

// ---------------- kernel 2: type-conditional token fill ----------------
__global__ void fill_tokens(const int* __restrict__ samples, float* __restrict__ tok) {
  int i = blockIdx.x * blockDim.x + threadIdx.x;   // i = b*D + d  (tokens.T layout)
  if (i >= Bz * Dz) return;
  int b = i / Dz;
  int d = i - b * Dz;
  int sv = samples[d * Bz + b];
  int s0 = samples[b];                              // field 0 sample for this batch
  tok[i] = (float)((d == 0 || s0 == NOTE) ? sv : 0);
}

// ---------------- entry point ----------------
extern "C" void kernel_launch(void* const* d_in, const int* in_sizes, int n_in,
                              void* d_out, int out_size, void* d_ws, size_t ws_size,
                              hipStream_t stream) {
  const float* logits = (const float*)d_in[0];   // [D,B,V] f32
  const float* u      = (const float*)d_in[1];   // [D,B,V] f32
  const int*   cur    = (const int*)d_in[2];     // [D,B]   i32

  float* out   = (float*)d_out;
  float* tok   = out;                             // first B*D floats: tokens.T
  float* probs = out + (size_t)Bz * Dz;           // then D*B*V floats: probs
  int*   samples = (int*)d_ws;                    // D*B ints of scratch

  decode_step<<<NBLK, BLK, 0, stream>>>(logits, u, cur, probs, samples);
  fill_tokens<<<(Bz * Dz + 255) / 256, 256, 0, stream>>>(samples, tok);
}